// GCNConv_47974784697087
// MI455X (gfx1250) — compile-verified
//
#include <hip/hip_runtime.h>

// GCNConv: agg = scatter_add(x[src] * norm_src[src], dst) * norm_dst;  out = agg @ W
// N = 100000 nodes, D = 64 feats, E = 1.6M edges. All fp32.

typedef float v2f __attribute__((ext_vector_type(2)));
typedef float v8f __attribute__((ext_vector_type(8)));

#define D 64

// ---------------- zero workspace ----------------
__global__ void gcn_zero_kernel(float* __restrict__ ws, long n) {
    long i = (long)blockIdx.x * blockDim.x + threadIdx.x;
    long stride = (long)gridDim.x * blockDim.x;
    for (; i < n; i += stride) ws[i] = 0.0f;
}

// ---------------- degree counts ----------------
__global__ void gcn_degree_kernel(const int* __restrict__ src, const int* __restrict__ dst,
                                  float* __restrict__ out_deg, float* __restrict__ in_deg, int E) {
    int e = blockIdx.x * blockDim.x + threadIdx.x;
    if (e < E) {
        atomicAdd(out_deg + src[e], 1.0f);
        atomicAdd(in_deg  + dst[e], 1.0f);
    }
}

// ---------------- deg -> clamp(deg,1)^-0.5 (in place) ----------------
__global__ void gcn_norm_kernel(float* __restrict__ deg_src, float* __restrict__ deg_dst, int N) {
    int i = blockIdx.x * blockDim.x + threadIdx.x;
    if (i < N) {
        deg_src[i] = rsqrtf(fmaxf(deg_src[i], 1.0f));
        deg_dst[i] = rsqrtf(fmaxf(deg_dst[i], 1.0f));
    }
}

// ---------------- edge scatter: agg[dst] += x[src] * norm_src[src] ----------------
// One thread per (edge, 4-feature chunk): E*16 threads, float4 read, 4 f32 atomics.
// agg working set (25.6MB) fits in the 192MB L2 where the atomic units live.
__global__ void gcn_scatter_kernel(const float* __restrict__ x, const int* __restrict__ src,
                                   const int* __restrict__ dst, const float* __restrict__ norm_src,
                                   float* __restrict__ agg, int E) {
    long tid = (long)blockIdx.x * blockDim.x + threadIdx.x;
    int e = (int)(tid >> 4);
    if (e >= E) return;
    int f = ((int)tid & 15) * 4;
    int s = src[e];
    int d = dst[e];
    float ns = norm_src[s];
    float4 v = *(const float4*)(x + (long)s * D + f);
    float* a = agg + (long)d * D + f;
    atomicAdd(a + 0, v.x * ns);
    atomicAdd(a + 1, v.y * ns);
    atomicAdd(a + 2, v.z * ns);
    atomicAdd(a + 3, v.w * ns);
}

// ---------------- out = (agg * norm_dst[:,None]) @ W via V_WMMA_F32_16X16X4_F32 ----------------
// One wave32 computes one 16x16 output tile; K=64 -> 16 WMMA steps.
// A frag (16x4 f32, 2 VGPR): lanes 0-15 = rows, K=kbase..kbase+1 ; lanes 16-31 same rows, K=kbase+2..3.
// B frag (4x16 f32, 2 VGPR): lanes 0-15 = cols with K rows kbase/kbase+1; lanes 16-31 K=+2/+3.
// C/D (16x16 f32, 8 VGPR): elem v -> row = v + 8*(lane>=16), col = lane%16.
__global__ void gcn_gemm_wmma_kernel(const float* __restrict__ agg,
                                     const float* __restrict__ norm_dst,
                                     const float* __restrict__ Wmat,
                                     float* __restrict__ out,
                                     int N, int numTiles) {
    int lane = threadIdx.x & 31;
    int wave = threadIdx.x >> 5;
    int w = blockIdx.x * (blockDim.x >> 5) + wave;
    if (w >= numTiles) return;           // whole-wave exit: EXEC all-1s at the WMMA

    int mtile = w >> 2;                  // N(=64)/16 = 4 column tiles
    int ntile = w & 3;
    int row0 = mtile * 16;
    int col0 = ntile * 16;

    int half = lane >> 4;                // 0: K pair {0,1}, 1: K pair {2,3} within each 4-K step
    int lm   = lane & 15;

    int m = row0 + lm;
    int mc = m < N ? m : (N - 1);        // clamp (no branch -> EXEC stays all-1s)
    float scale = (m < N) ? norm_dst[mc] : 0.0f;
    const float* arow = agg + (long)mc * D;

    v8f c = {};
    #pragma unroll
    for (int kk = 0; kk < 16; ++kk) {
        int kbase = kk * 4 + half * 2;
        float2 av = *(const float2*)(arow + kbase);   // A row-major: K,K+1 contiguous
        v2f a; a.x = av.x * scale; a.y = av.y * scale;
        v2f b;
        b.x = Wmat[(kbase + 0) * D + col0 + lm];      // W row-major [K][Nout]
        b.y = Wmat[(kbase + 1) * D + col0 + lm];
        // 8 args: (neg_a, A, neg_b, B, c_mod, C, reuse_a, reuse_b)
        c = __builtin_amdgcn_wmma_f32_16x16x4_f32(false, a, false, b, (short)0, c, false, false);
    }

    #pragma unroll
    for (int v = 0; v < 8; ++v) {
        int row = row0 + v + half * 8;
        if (row < N)
            out[(long)row * D + col0 + lm] = c[v];
    }
}

extern "C" void kernel_launch(void* const* d_in, const int* in_sizes, int n_in,
                              void* d_out, int out_size, void* d_ws, size_t ws_size,
                              hipStream_t stream) {
    const float* x    = (const float*)d_in[0];   // [N,64]
    const float* Wmat = (const float*)d_in[1];   // [64,64]
    const int*   src  = (const int*)d_in[2];     // [E]
    const int*   dst  = (const int*)d_in[3];     // [E]
    float* out = (float*)d_out;                  // [N,64]

    int N = in_sizes[0] / D;
    int E = in_sizes[2];

    // Workspace layout: [norm_src: N][norm_dst: N][agg: N*D]  (~26.4 MB for N=100000)
    float* ws_norm_src = (float*)d_ws;
    float* ws_norm_dst = ws_norm_src + N;
    float* ws_agg      = ws_norm_dst + N;

    long nzero = (long)N * (D + 2);
    {
        int blocks = (int)((nzero + 255) / 256);
        if (blocks > 65535) blocks = 65535;   // grid-stride covers the rest
        gcn_zero_kernel<<<blocks, 256, 0, stream>>>((float*)d_ws, nzero);
    }

    gcn_degree_kernel<<<(E + 255) / 256, 256, 0, stream>>>(src, dst, ws_norm_src, ws_norm_dst, E);

    gcn_norm_kernel<<<(N + 255) / 256, 256, 0, stream>>>(ws_norm_src, ws_norm_dst, N);

    {
        long work = (long)E * 16;             // (edge, 4-feat chunk)
        int blocks = (int)((work + 255) / 256);
        gcn_scatter_kernel<<<blocks, 256, 0, stream>>>(x, src, dst, ws_norm_src, ws_agg, E);
    }

    {
        int mtiles = (N + 15) / 16;
        int numTiles = mtiles * 4;            // 64/16 output-column tiles
        int wavesPerBlock = 8;                // 256 threads = 8 wave32
        int blocks = (numTiles + wavesPerBlock - 1) / wavesPerBlock;
        gcn_gemm_wmma_kernel<<<blocks, 256, 0, stream>>>(ws_agg, ws_norm_dst, Wmat, out, N, numTiles);
    }
}